// MoEHead_38379827757765
// MI455X (gfx1250) — compile-verified
//
#include <hip/hip_runtime.h>
#include <hip/hip_bf16.h>
#include <math.h>
#include <stdint.h>

#define SLEN   512
#define DIMV   1024
#define VOCABV 32000
#define EXP    8
#define TOPK   2
#define DFFV   2730
#define DFF2V  5460
#define NPAIR  1024          // SLEN * TOPK
#define MAXTILES 72          // 1024/16 + 8 padding tiles (worst case)
#define KC     64            // K chunk staged in LDS
#define LDA    68            // padded LDS row stride (floats): 64 data + 4 pad

typedef float v2f __attribute__((ext_vector_type(2)));
typedef float v8f __attribute__((ext_vector_type(8)));
typedef unsigned int u32x4 __attribute__((ext_vector_type(4)));
typedef int i32x8 __attribute__((ext_vector_type(8)));
typedef int i32x4 __attribute__((ext_vector_type(4)));

#if __has_builtin(__builtin_amdgcn_tensor_load_to_lds)
#define HAVE_TDM 1
#else
#define HAVE_TDM 0
#endif

__device__ __forceinline__ float gelu_exact(float g) {
    return 0.5f * g * (1.0f + erff(g * 0.70710678118654752440f));
}

__device__ __forceinline__ unsigned lds_off(const void* p) {
    // AS(3) -> generic is {SHARED_BASE, offset}; low 32 bits = LDS byte offset
    return (unsigned)(uintptr_t)p;
}

// one wave-level async copy of 128B per lane (8 x b128), LDS and global advance together
#define ASYNC_CHUNK(ldsA, gOff, basePtr)                                                        \
    do {                                                                                        \
        asm volatile("global_load_async_to_lds_b128 %0, %1, %2 offset:0"                        \
                     :: "v"(ldsA), "v"(gOff), "s"(basePtr) : "memory");                         \
        asm volatile("global_load_async_to_lds_b128 %0, %1, %2 offset:16"                       \
                     :: "v"(ldsA), "v"(gOff), "s"(basePtr) : "memory");                         \
        asm volatile("global_load_async_to_lds_b128 %0, %1, %2 offset:32"                       \
                     :: "v"(ldsA), "v"(gOff), "s"(basePtr) : "memory");                         \
        asm volatile("global_load_async_to_lds_b128 %0, %1, %2 offset:48"                       \
                     :: "v"(ldsA), "v"(gOff), "s"(basePtr) : "memory");                         \
        asm volatile("global_load_async_to_lds_b128 %0, %1, %2 offset:64"                       \
                     :: "v"(ldsA), "v"(gOff), "s"(basePtr) : "memory");                         \
        asm volatile("global_load_async_to_lds_b128 %0, %1, %2 offset:80"                       \
                     :: "v"(ldsA), "v"(gOff), "s"(basePtr) : "memory");                         \
        asm volatile("global_load_async_to_lds_b128 %0, %1, %2 offset:96"                       \
                     :: "v"(ldsA), "v"(gOff), "s"(basePtr) : "memory");                         \
        asm volatile("global_load_async_to_lds_b128 %0, %1, %2 offset:112"                      \
                     :: "v"(ldsA), "v"(gOff), "s"(basePtr) : "memory");                         \
    } while (0)

#define WAIT_ASYNC_8() asm volatile("s_wait_asynccnt 8" ::: "memory")
#define WAIT_ASYNC_0() asm volatile("s_wait_asynccnt 0" ::: "memory")

#if HAVE_TDM
// TDM: move a 2D tile (nLines lines x 64 floats, line stride strideElems floats)
// from global memory into LDS laid out with LDA=68-float padded rows.
__device__ __forceinline__ void tdm_load_tile(unsigned ldsAddr, unsigned long long gaddr,
                                              unsigned nLines, unsigned strideElems) {
    union { unsigned long long q[2]; u32x4 v; } g0;
    union { unsigned long long q[4]; i32x8 v; } g1;
    // group0: count=1 | lds_addr<<32 ; global_addr[56:0] | type=2<<126
    g0.q[0] = 1ULL | ((unsigned long long)ldsAddr << 32);
    g0.q[1] = (gaddr & ((1ULL << 57) - 1)) | (2ULL << 62);
    // group1: data_size=2 (4B) | pad_enable | pad_interval=5 (64 DW) | pad_amount=3 (4 DW)
    const unsigned long long td0 = 0x40000000ULL;   // huge tensor dims: no clipping
    const unsigned long long td1 = 0x40000000ULL;
    g1.q[0] = (unsigned long long)((2u << 16) | (1u << 20) | (5u << 22) | (3u << 25))
            | ((td0 & 0xFFFFULL) << 48);                         // tensor_dim0[15:0] @ 63:48
    g1.q[1] = (td0 >> 16)                                        // tensor_dim0[31:16] @ 79:64
            | (td1 << 16)                                        // tensor_dim1 @ 111:80
            | (64ULL << 48);                                     // tile_dim0 @ 127:112
    g1.q[2] = (unsigned long long)nLines                          // tile_dim1 @ 143:128
            | ((unsigned long long)strideElems << 32);            // stride0[31:0] @ 191:160
    g1.q[3] = 0ULL;                                               // stride0 hi = 0, dim1_stride = 0
    i32x4 z4 = {0, 0, 0, 0};
    i32x8 z8 = {0, 0, 0, 0, 0, 0, 0, 0};
    __builtin_amdgcn_tensor_load_to_lds(g0.v, g1.v, z4, z4, z8, 0);
}
#define WAIT_TENSOR_1() __builtin_amdgcn_s_wait_tensorcnt((short)1)
#define WAIT_TENSOR_0() __builtin_amdgcn_s_wait_tensorcnt((short)0)
#endif

// ---------------------------------------------------------------- router
__global__ void router_kernel(const float* __restrict__ x, const float* __restrict__ Wr,
                              int* __restrict__ ids, float* __restrict__ gate) {
    int s = blockIdx.x * blockDim.x + threadIdx.x;
    if (s >= SLEN) return;
    const float* xr = x + (size_t)s * DIMV;
    float sc[EXP];
#pragma unroll
    for (int e = 0; e < EXP; ++e) sc[e] = 0.f;
    for (int d = 0; d < DIMV; ++d) {
        float xv = xr[d];
#pragma unroll
        for (int e = 0; e < EXP; ++e) sc[e] = fmaf(xv, Wr[e * DIMV + d], sc[e]);
    }
    int i0 = 0;
#pragma unroll
    for (int e = 1; e < EXP; ++e) if (sc[e] > sc[i0]) i0 = e;
    int i1 = (i0 == 0) ? 1 : 0;
#pragma unroll
    for (int e = 0; e < EXP; ++e) if (e != i0 && sc[e] > sc[i1]) i1 = e;
    float s0 = sc[i0], s1 = sc[i1];
    float mx = fmaxf(s0, s1);
    float l  = mx + logf(expf(s0 - mx) + expf(s1 - mx));
    ids[s * 2 + 0]  = i0;      ids[s * 2 + 1]  = i1;
    gate[s * 2 + 0] = s0 - l;  gate[s * 2 + 1] = s1 - l;
}

// ---------------------------------------------------------------- bucket by expert
__global__ void bucket_kernel(const int* __restrict__ ids, int* __restrict__ sortedPair,
                              int* __restrict__ tileExpert, int* __restrict__ tileRow,
                              int* __restrict__ tileValid) {
    __shared__ int cnt[EXP], base[EXP], off[EXP];
    int t = threadIdx.x;                // 0..NPAIR-1
    if (t < EXP) cnt[t] = 0;
    __syncthreads();
    int e = ids[t];
    atomicAdd(&cnt[e], 1);
    __syncthreads();
    if (t == 0) {
        int acc = 0;
        for (int i = 0; i < EXP; ++i) { base[i] = acc; off[i] = acc; acc += cnt[i]; }
    }
    __syncthreads();
    int pos = atomicAdd(&off[e], 1);
    sortedPair[pos] = t;
    __syncthreads();
    if (t == 0) {
        int nt = 0;
        for (int i = 0; i < EXP; ++i)
            for (int r = 0; r < cnt[i]; r += 16) {
                tileExpert[nt] = i;
                tileRow[nt]    = base[i] + r;
                tileValid[nt]  = min(16, cnt[i] - r);
                ++nt;
            }
        for (; nt < MAXTILES; ++nt) { tileExpert[nt] = -1; tileRow[nt] = 0; tileValid[nt] = 0; }
    }
}

// ---------------------------------------------------------------- wmma helper
__device__ __forceinline__ v8f wmma_k64(const float* __restrict__ pa,
                                        const float* __restrict__ pb,
                                        int kh, v8f c) {
#pragma unroll
    for (int k0 = 0; k0 < KC; k0 += 4) {
        int ko = k0 + 2 * kh;
        v2f a, b;
        a.x = pa[ko]; a.y = pa[ko + 1];
        b.x = pb[ko]; b.y = pb[ko + 1];
        c = __builtin_amdgcn_wmma_f32_16x16x4_f32(false, a, false, b, (short)0, c, false, false);
    }
    return c;
}

// ---------------------------------------------------------------- Wup GEMM + GeGLU
__global__ __launch_bounds__(128)
void up_gemm_kernel(const float* __restrict__ x, const float* __restrict__ Wup,
                    const float* __restrict__ bup, const int* __restrict__ sortedPair,
                    const int* __restrict__ tileExpert, const int* __restrict__ tileRow,
                    const int* __restrict__ tileValid, float* __restrict__ hmid) {
    __shared__ float sA[2][16 * LDA];
    __shared__ float sB[2][4][16 * LDA];
    __shared__ float sC[4][16 * 17];
    __shared__ int   sPair[16];
    __shared__ int   sValid;

    int e = tileExpert[blockIdx.y];
    if (e < 0) return;
    int t = threadIdx.x;
    if (t == 0) sValid = tileValid[blockIdx.y];
    if (t < 16) sPair[t] = (t < tileValid[blockIdx.y]) ? sortedPair[tileRow[blockIdx.y] + t] : -1;
    __syncthreads();

    int j0 = blockIdx.x * 32;
    int w = t >> 5, lane = t & 31, lm = lane & 15, kh = lane >> 4;

    v8f c;
#pragma unroll
    for (int i = 0; i < 8; ++i) c[i] = 0.f;

    int arow = t >> 3, ac0 = (t & 7) * 8;    // A: 16 rows x 64, 8 floats/thread
    int brow = t >> 1, bc0 = (t & 1) * 32;   // B: 64 rows x 64, 32 floats/thread (128B async)
    int bw = brow >> 4, bn = brow & 15;
    int jcol = j0 + ((bw & 1) << 4) + bn;
    bool bval = jcol < DFFV;
    int grow = ((bw >= 2) ? DFFV : 0) + jcol;
    int bByte = bval ? ((e * DFF2V + grow) * DIMV + bc0) * 4 : 0;   // redirect invalid rows
    unsigned long long wBase = (unsigned long long)Wup;
    int apr  = sPair[arow];
    bool aOk = (apr >= 0);
    const float* aSafe = aOk ? (x + (size_t)(apr >> 1) * DIMV) : x;

    auto stageA = [&](int kc, int buf) {
#pragma unroll
        for (int j = 0; j < 8; ++j) {
            float v = aSafe[kc + ac0 + j];            // always in-bounds
            sA[buf][arow * LDA + ac0 + j] = aOk ? v : 0.f;
        }
    };
    auto stageB = [&](int kc, int buf) {
        unsigned lds = lds_off(&sB[buf][bw][bn * LDA + bc0]);
        int voff = bByte + kc * 4;
        ASYNC_CHUNK(lds, voff, wBase);
    };

    const int NCH = DIMV / KC;                        // 16
    stageB(0, 0);
    stageA(0, 0);
    for (int i = 0; i < NCH; ++i) {
        int cur = i & 1;
        if (i + 1 < NCH) {
            stageB((i + 1) * KC, cur ^ 1);
            stageA((i + 1) * KC, cur ^ 1);
            WAIT_ASYNC_8();
        } else {
            WAIT_ASYNC_0();
        }
        __syncthreads();
        c = wmma_k64(&sA[cur][lm * LDA], &sB[cur][w][lm * LDA], kh, c);
        __syncthreads();
    }

#pragma unroll
    for (int i = 0; i < 8; ++i) {
        int m = i + ((lane >= 16) ? 8 : 0);
        sC[w][m * 17 + lm] = c[i];
    }
    __syncthreads();
    int valid = sValid;
#pragma unroll
    for (int oi = 0; oi < 4; ++oi) {
        int idx = oi * 128 + t;          // 0..511 -> 16 rows x 32 cols
        int m = idx >> 5, col = idx & 31;
        int wh = col >> 4, n = col & 15;
        int j = j0 + col;
        if (m < valid && j < DFFV) {
            float h = sC[wh][m * 17 + n]     + bup[(size_t)e * DFF2V + j];
            float g = sC[2 + wh][m * 17 + n] + bup[(size_t)e * DFF2V + DFFV + j];
            int pr = sPair[m];
            hmid[(size_t)pr * DFFV + j] = h * gelu_exact(g);
        }
    }
}

// ---------------------------------------------------------------- Wdown GEMM + residual
__global__ __launch_bounds__(128)
void down_gemm_kernel(const float* __restrict__ x, const float* __restrict__ hmid,
                      const float* __restrict__ Wdown, const int* __restrict__ sortedPair,
                      const int* __restrict__ tileExpert, const int* __restrict__ tileRow,
                      const int* __restrict__ tileValid, float* __restrict__ h2) {
    __shared__ float sA[2][16 * LDA];
    __shared__ float sB[2][4][16 * LDA];
    __shared__ int   sPair[16];
    __shared__ int   sValid;

    int e = tileExpert[blockIdx.y];
    if (e < 0) return;
    int t = threadIdx.x;
    if (t == 0) sValid = tileValid[blockIdx.y];
    if (t < 16) sPair[t] = (t < tileValid[blockIdx.y]) ? sortedPair[tileRow[blockIdx.y] + t] : -1;
    __syncthreads();

    int w = t >> 5, lane = t & 31, lm = lane & 15, kh = lane >> 4;
    int colBase = blockIdx.x * 64 + w * 16;

    v8f c;
#pragma unroll
    for (int i = 0; i < 8; ++i) c[i] = 0.f;

    int arow = t >> 3, ac0 = (t & 7) * 8;
    int brow = t >> 1, bc0 = (t & 1) * 32;
    int bw = brow >> 4, bn = brow & 15;
    int gcol = blockIdx.x * 64 + bw * 16 + bn;        // always < DIMV
    int rowIdx = e * DIMV + gcol;
    int bByte = (rowIdx * DFFV + bc0) * 4;
    unsigned long long wBase = (unsigned long long)Wdown;
    const float* bRow = Wdown + (size_t)rowIdx * DFFV;
    int apr  = sPair[arow];
    bool aOk = (apr >= 0);
    const float* aSafe = aOk ? (hmid + (size_t)apr * DFFV) : hmid;

    auto stageA = [&](int kc, int buf) {
#pragma unroll
        for (int j = 0; j < 8; ++j) {
            int k = kc + ac0 + j;
            int ks = (k < DFFV) ? k : (DFFV - 1);
            float v = aSafe[ks];
            sA[buf][arow * LDA + ac0 + j] = (aOk && k < DFFV) ? v : 0.f;
        }
    };
    auto stageBasync = [&](int kc, int buf) {         // full in-row chunk
        unsigned lds = lds_off(&sB[buf][bw][bn * LDA + bc0]);
        int voff = bByte + kc * 4;
        ASYNC_CHUNK(lds, voff, wBase);
    };
    auto stageBtail = [&](int kc, int buf) {          // K tail, exact zero-pad, in-bounds
#pragma unroll
        for (int j = 0; j < 32; ++j) {
            int k = kc + bc0 + j;
            int ks = (k < DFFV) ? k : (DFFV - 1);
            float v = bRow[ks];
            sB[buf][bw][bn * LDA + bc0 + j] = (k < DFFV) ? v : 0.f;
        }
    };

    const int NCH = (DFFV + KC - 1) / KC;             // 43, last chunk is tail
    stageBasync(0, 0);
    stageA(0, 0);
    for (int i = 0; i < NCH; ++i) {
        int cur = i & 1;
        if (i + 1 < NCH && (i + 1) * KC + KC <= DFFV) {
            stageBasync((i + 1) * KC, cur ^ 1);
            stageA((i + 1) * KC, cur ^ 1);
            WAIT_ASYNC_8();
        } else if (i + 1 < NCH) {
            stageBtail((i + 1) * KC, cur ^ 1);
            stageA((i + 1) * KC, cur ^ 1);
            WAIT_ASYNC_0();
        } else {
            WAIT_ASYNC_0();
        }
        __syncthreads();
        c = wmma_k64(&sA[cur][lm * LDA], &sB[cur][w][lm * LDA], kh, c);
        __syncthreads();
    }

    int valid = sValid;
#pragma unroll
    for (int i = 0; i < 8; ++i) {
        int m = i + ((lane >= 16) ? 8 : 0);
        if (m < valid) {
            int pr = sPair[m];
            int tok = pr >> 1;
            int col = colBase + lm;
            h2[(size_t)pr * DIMV + col] = c[i] + x[(size_t)tok * DIMV + col];
        }
    }
}

// ---------------------------------------------------------------- Wproj GEMM -> logits (TDM B-path)
__global__ __launch_bounds__(128)
void proj_gemm_kernel(const float* __restrict__ h2, const float* __restrict__ Wproj,
                      const int* __restrict__ sortedPair,
                      const int* __restrict__ tileExpert, const int* __restrict__ tileRow,
                      const int* __restrict__ tileValid, float* __restrict__ logits) {
    __shared__ float sA[2][16 * LDA];
    __shared__ float sB[2][4][16 * LDA];
    __shared__ int   sPair[16];
    __shared__ int   sValid;

    int e = tileExpert[blockIdx.y];
    if (e < 0) return;
    int t = threadIdx.x;
    if (t == 0) sValid = tileValid[blockIdx.y];
    if (t < 16) sPair[t] = (t < tileValid[blockIdx.y]) ? sortedPair[tileRow[blockIdx.y] + t] : -1;
    __syncthreads();

    int w = t >> 5, lane = t & 31, lm = lane & 15, kh = lane >> 4;
    int colBase = blockIdx.x * 64 + w * 16;           // always < VOCABV

    v8f c;
#pragma unroll
    for (int i = 0; i < 8; ++i) c[i] = 0.f;

    int arow = t >> 3, ac0 = (t & 7) * 8;
    int apr  = sPair[arow];
    bool aOk = (apr >= 0);
    const float* aSafe = aOk ? (h2 + (size_t)apr * DIMV) : h2;

    auto stageA = [&](int kc, int buf) {
#pragma unroll
        for (int j = 0; j < 8; ++j) {
            float v = aSafe[kc + ac0 + j];
            sA[buf][arow * LDA + ac0 + j] = aOk ? v : 0.f;
        }
    };

#if HAVE_TDM
    // one TDM op per block per chunk: 64 weight rows x 64 floats, stride DIMV, into padded LDS
    unsigned long long rowBase = (unsigned long long)Wproj
                               + ((unsigned long long)(e * VOCABV + blockIdx.x * 64) * DIMV) * 4ULL;
    auto stageB = [&](int kc, int buf) {
        if (t < 32) {   // wave 0 drives the block's tensor DMA
            tdm_load_tile(lds_off(&sB[buf][0][0]), rowBase + (unsigned long long)kc * 4ULL,
                          64u, (unsigned)DIMV);
        }
    };
#else
    int brow = t >> 1, bc0 = (t & 1) * 32;
    int bw = brow >> 4, bn = brow & 15;
    int gcol = blockIdx.x * 64 + bw * 16 + bn;
    int bByte = ((e * VOCABV + gcol) * DIMV + bc0) * 4;   // max ~1.05e9 < 2^31
    unsigned long long wBase = (unsigned long long)Wproj;
    auto stageB = [&](int kc, int buf) {
        unsigned lds = lds_off(&sB[buf][bw][bn * LDA + bc0]);
        int voff = bByte + kc * 4;
        ASYNC_CHUNK(lds, voff, wBase);
    };
#endif

    const int NCH = DIMV / KC;                        // 16
    stageB(0, 0);
    stageA(0, 0);
    for (int i = 0; i < NCH; ++i) {
        int cur = i & 1;
        if (i + 1 < NCH) {
            stageB((i + 1) * KC, cur ^ 1);
            stageA((i + 1) * KC, cur ^ 1);
#if HAVE_TDM
            WAIT_TENSOR_1();
#else
            WAIT_ASYNC_8();
#endif
        } else {
#if HAVE_TDM
            WAIT_TENSOR_0();
#else
            WAIT_ASYNC_0();
#endif
        }
        __syncthreads();
        c = wmma_k64(&sA[cur][lm * LDA], &sB[cur][w][lm * LDA], kh, c);
        __syncthreads();
    }

    int valid = sValid;
#pragma unroll
    for (int i = 0; i < 8; ++i) {
        int m = i + ((lane >= 16) ? 8 : 0);
        if (m < valid) {
            int pr = sPair[m];
            logits[(size_t)pr * VOCABV + colBase + lm] = c[i];
        }
    }
}

// ---------------------------------------------------------------- per-row logsumexp
__global__ void lse_kernel(const float* __restrict__ logits, float* __restrict__ lse) {
    __shared__ float red[256];
    int t = threadIdx.x;
    const float* lr = logits + (size_t)blockIdx.x * VOCABV;
    float mx = -3.4e38f;
    for (int v = t; v < VOCABV; v += 256) mx = fmaxf(mx, lr[v]);
    red[t] = mx; __syncthreads();
    for (int s = 128; s > 0; s >>= 1) { if (t < s) red[t] = fmaxf(red[t], red[t + s]); __syncthreads(); }
    mx = red[0]; __syncthreads();
    float sum = 0.f;
    for (int v = t; v < VOCABV; v += 256) sum += expf(lr[v] - mx);
    red[t] = sum; __syncthreads();
    for (int s = 128; s > 0; s >>= 1) { if (t < s) red[t] += red[t + s]; __syncthreads(); }
    if (t == 0) lse[blockIdx.x] = mx + logf(red[0]);
}

// ---------------------------------------------------------------- top-2 combine
__global__ void combine_kernel(const float* __restrict__ logits, const float* __restrict__ lse,
                               const float* __restrict__ gate, float* __restrict__ out) {
    long long idx = (long long)blockIdx.x * 256 + threadIdx.x;
    if (idx >= (long long)SLEN * VOCABV) return;
    int s = (int)(idx / VOCABV);
    int v = (int)(idx - (long long)s * VOCABV);
    float t0 = logits[(size_t)(2 * s + 0) * VOCABV + v] - lse[2 * s + 0] + gate[2 * s + 0];
    float t1 = logits[(size_t)(2 * s + 1) * VOCABV + v] - lse[2 * s + 1] + gate[2 * s + 1];
    float m = fmaxf(t0, t1);
    out[idx] = m + logf(expf(t0 - m) + expf(t1 - m));
}

// ---------------------------------------------------------------- host launch
extern "C" void kernel_launch(void* const* d_in, const int* in_sizes, int n_in,
                              void* d_out, int out_size, void* d_ws, size_t ws_size,
                              hipStream_t stream) {
    const float* x     = (const float*)d_in[0];
    const float* Wr    = (const float*)d_in[1];
    const float* Wup   = (const float*)d_in[2];
    const float* bup   = (const float*)d_in[3];
    const float* Wdown = (const float*)d_in[4];
    const float* Wproj = (const float*)d_in[5];
    float* out = (float*)d_out;

    char* ws = (char*)d_ws;
    int*   ids        = (int*)  (ws + 0);
    float* gate       = (float*)(ws + 4096);
    int*   sortedPair = (int*)  (ws + 8192);
    int*   tileExpert = (int*)  (ws + 12288);
    int*   tileRow    = (int*)  (ws + 12800);
    int*   tileValid  = (int*)  (ws + 13312);
    size_t o = 16384;
    float* hmid   = (float*)(ws + o);  o += (size_t)NPAIR * DFFV * 4;
    float* h2     = (float*)(ws + o);  o += (size_t)NPAIR * DIMV * 4;
    float* lse    = (float*)(ws + o);  o += 4096;
    float* logits = (float*)(ws + o);  // NPAIR * VOCABV * 4 = 131 MB

    router_kernel<<<2, 256, 0, stream>>>(x, Wr, ids, gate);
    bucket_kernel<<<1, NPAIR, 0, stream>>>(ids, sortedPair, tileExpert, tileRow, tileValid);
    up_gemm_kernel<<<dim3((DFFV + 31) / 32, MAXTILES), 128, 0, stream>>>(
        x, Wup, bup, sortedPair, tileExpert, tileRow, tileValid, hmid);
    down_gemm_kernel<<<dim3(DIMV / 64, MAXTILES), 128, 0, stream>>>(
        x, hmid, Wdown, sortedPair, tileExpert, tileRow, tileValid, h2);
    proj_gemm_kernel<<<dim3(VOCABV / 64, MAXTILES), 128, 0, stream>>>(
        h2, Wproj, sortedPair, tileExpert, tileRow, tileValid, logits);
    lse_kernel<<<NPAIR, 256, 0, stream>>>(logits, lse);
    combine_kernel<<<(int)(((long long)SLEN * VOCABV + 255) / 256), 256, 0, stream>>>(
        logits, lse, gate, out);
}